// MyEdgeConv_30820685316843
// MI455X (gfx1250) — compile-verified
//
#include <hip/hip_runtime.h>

typedef __attribute__((ext_vector_type(2))) float v2f;
typedef __attribute__((ext_vector_type(8))) float v8f;

#define N_EDGES  600000
#define HF       128          // H
#define IN_F     384          // 3H
#define MID_F    256          // 2H
#define TILE_E   64           // edges per workgroup
#define A_STRIDE 388          // 384 + 4 pad  (388 % 64 == 4 -> conflict-free frag loads)
#define H_STRIDE 260          // 256 + 4 pad  (260 % 64 == 4)

__global__ __launch_bounds__(256, 1)
void edge_mlp_wmma_f32(const float* __restrict__ x,
                       const int*   __restrict__ eidx,
                       const float* __restrict__ ea,
                       const float* __restrict__ W1,
                       const float* __restrict__ b1,
                       const float* __restrict__ W2,
                       const float* __restrict__ b2,
                       float* __restrict__ out)
{
    __shared__ float As[TILE_E * A_STRIDE];   // 99,328 B
    __shared__ float Hs[TILE_E * H_STRIDE];   // 66,560 B
    __shared__ int   rowIdx[TILE_E];
    __shared__ int   colIdx[TILE_E];

    const int tid  = threadIdx.x;
    const int wave = tid >> 5;
    const int lane = tid & 31;
    const int m    = lane & 15;   // row (A) / col (B) within 16
    const int half = lane >> 4;   // K sub-offset selector (0 -> k0..k0+1, 1 -> k0+2..k0+3)
    const int e0   = blockIdx.x * TILE_E;

    if (tid < TILE_E) {
        rowIdx[tid] = eidx[e0 + tid];
        colIdx[tid] = eidx[N_EDGES + e0 + tid];
    }
    __syncthreads();

    // ---- stage A = [x[row] | x[col] | edge_attr] : 64 x 384 fp32 -> LDS ----
    #pragma unroll
    for (int i = 0; i < (TILE_E * (IN_F / 4)) / 256; ++i) {   // 24 float4 per thread
        int s   = tid + i * 256;
        int e   = s / (IN_F / 4);          // /96
        int seg = s - e * (IN_F / 4);
        int f4  = seg * 4;                 // float offset within the 384-wide row
        const float* src;
        if (seg < 32)      src = x  + (size_t)rowIdx[e] * HF + f4;
        else if (seg < 64) src = x  + (size_t)colIdx[e] * HF + (f4 - 128);
        else               src = ea + (size_t)(e0 + e) * HF + (f4 - 256);
        *(float4*)(&As[e * A_STRIDE + f4]) = *(const float4*)src;
    }
    __syncthreads();

    // ---- GEMM1: h[64,256] = relu(A[64,384] * W1^T + b1) ----
    // Each wave: ALL 4 row tiles x 2 col tiles  ->  B fragments reused 4x,
    // W1 streamed exactly once per workgroup.
    {
        const int ct0 = wave * 2;          // col tiles {2w, 2w+1} of 16
        v8f acc[4][2] = {};

        #pragma unroll 2
        for (int k0 = 0; k0 < IN_F; k0 += 4) {
            v2f a[4];
            #pragma unroll
            for (int rt = 0; rt < 4; ++rt)
                a[rt] = *(const v2f*)(&As[(rt * 16 + m) * A_STRIDE + k0 + 2 * half]);
            v2f b[2];
            #pragma unroll
            for (int j = 0; j < 2; ++j) {
                int ncol = ((ct0 + j) << 4) + m;
                b[j] = *(const v2f*)(W1 + (size_t)ncol * IN_F + k0 + 2 * half);
            }
            #pragma unroll
            for (int rt = 0; rt < 4; ++rt) {
                #pragma unroll
                for (int j = 0; j < 2; ++j) {
                    acc[rt][j] = __builtin_amdgcn_wmma_f32_16x16x4_f32(
                        false, a[rt], false, b[j], (short)0, acc[rt][j], false, false);
                }
            }
        }
        #pragma unroll
        for (int j = 0; j < 2; ++j) {
            int   ncol = ((ct0 + j) << 4) + m;
            float bias = b1[ncol];
            #pragma unroll
            for (int rt = 0; rt < 4; ++rt) {
                #pragma unroll
                for (int v = 0; v < 8; ++v) {
                    float hv = acc[rt][j][v] + bias;
                    hv = hv > 0.f ? hv : 0.f;
                    Hs[(rt * 16 + v + 8 * half) * H_STRIDE + ncol] = hv;
                }
            }
        }
    }
    __syncthreads();

    // ---- GEMM2: out[64,128] = h[64,256] * W2^T + b2 ----
    // Each wave: ALL 4 row tiles x 1 col tile (8 waves cover 128 outputs).
    {
        const int ct   = wave;             // col tile 0..7
        const int ncol = (ct << 4) + m;
        v8f acc[4] = {};

        #pragma unroll 2
        for (int g0 = 0; g0 < MID_F; g0 += 4) {
            v2f a[4];
            #pragma unroll
            for (int rt = 0; rt < 4; ++rt)
                a[rt] = *(const v2f*)(&Hs[(rt * 16 + m) * H_STRIDE + g0 + 2 * half]);
            v2f b = *(const v2f*)(W2 + (size_t)ncol * MID_F + g0 + 2 * half);
            #pragma unroll
            for (int rt = 0; rt < 4; ++rt) {
                acc[rt] = __builtin_amdgcn_wmma_f32_16x16x4_f32(
                    false, a[rt], false, b, (short)0, acc[rt], false, false);
            }
        }
        float bias = b2[ncol];
        #pragma unroll
        for (int rt = 0; rt < 4; ++rt) {
            #pragma unroll
            for (int v = 0; v < 8; ++v) {
                out[(size_t)(e0 + rt * 16 + v + 8 * half) * HF + ncol] = acc[rt][v] + bias;
            }
        }
    }
}

extern "C" void kernel_launch(void* const* d_in, const int* in_sizes, int n_in,
                              void* d_out, int out_size, void* d_ws, size_t ws_size,
                              hipStream_t stream) {
    const float* x   = (const float*)d_in[0];
    const int*   ei  = (const int*)  d_in[1];
    const float* ea  = (const float*)d_in[2];
    const float* W1  = (const float*)d_in[3];
    const float* b1  = (const float*)d_in[4];
    const float* W2  = (const float*)d_in[5];
    const float* b2  = (const float*)d_in[6];
    float*       out = (float*)d_out;

    dim3 grid(N_EDGES / TILE_E);   // 9375
    dim3 block(256);
    edge_mlp_wmma_f32<<<grid, block, 0, stream>>>(x, ei, ea, W1, b1, W2, b2, out);
}